// Compute_Gate_78735340470740
// MI455X (gfx1250) — compile-verified
//
#include <hip/hip_runtime.h>
#include <math.h>

// GEMM view:  M = B = 4096 (batch), N = C*2H = 32768, K = IN + H = 2048.
// A = [X | Hs] (row stride 1024 each half), B[n][k] = {Wx,Wh}[n][k] (k-contig).
// Memory-bound: ~840 MB @ 23.3 TB/s ~ 36 us >> compute at any precision, so
// stay f32 (V_WMMA_F32_16X16X4_F32) and optimize streaming:
//   - async global->LDS DMA (ASYNCcnt) with double-buffered LDS
//   - grid.x = M-tiles so each 1 MB weight tile is L2-resident across 32 blocks
//   - non-temporal output stores (write-once 537 MB must not evict L2)

typedef __attribute__((ext_vector_type(2))) float v2f;
typedef __attribute__((ext_vector_type(8))) float v8f;
typedef __attribute__((ext_vector_type(4))) int   v4i;

#define AS1 __attribute__((address_space(1)))
#define AS3 __attribute__((address_space(3)))

#define IN_DIM 1024
#define H_DIM  1024
#define C_DIM  16
#define B_DIM  4096
#define KDIM   2048
#define NDIM   32768

#define BM 128
#define BN 128
#define BK 32
// 36 floats = 144 B rows: 16B-aligned (b128 LDS ok) and i*36 mod 64 distinct
// for i=0..15 -> conflict-free b64 fragment reads.
#define LSTR 36
#define TILE_F (BM * LSTR)

#if __has_builtin(__builtin_amdgcn_global_load_async_to_lds_b128)
#define HAVE_ASYNC_LDS 1
#else
#define HAVE_ASYNC_LDS 0
#endif

__device__ __forceinline__ void wait_async_all() {
#if HAVE_ASYNC_LDS
#if __has_builtin(__builtin_amdgcn_s_wait_asynccnt)
    __builtin_amdgcn_s_wait_asynccnt(0);
#else
    asm volatile("s_wait_asynccnt 0x0" ::: "memory");
#endif
#endif
}

__device__ __forceinline__ float fast_tanh(float x) {
#if __has_builtin(__builtin_amdgcn_tanhf)
    return __builtin_amdgcn_tanhf(x);
#elif __has_builtin(__builtin_amdgcn_tanh_f32)
    return __builtin_amdgcn_tanh_f32(x);
#else
    return tanhf(x);
#endif
}

// Stage one (BM x BK) A tile and (BN x BK) B tile into LDS buffers.
__device__ __forceinline__ void stage_tiles(int kt, float* __restrict__ AsBuf,
                                            float* __restrict__ BsBuf,
                                            const float* __restrict__ X,
                                            const float* __restrict__ Hs,
                                            const float* __restrict__ Wx,
                                            const float* __restrict__ Wh,
                                            int m0, int n0, int lrow, int lcol) {
    const bool first = (kt < IN_DIM);
    const int ko = first ? kt : (kt - IN_DIM);
    const float* aSrc = (first ? X : Hs) + ko;
    const float* wSrc = (first ? Wx : Wh) + ko;
#if HAVE_ASYNC_LDS
#pragma unroll
    for (int p = 0; p < 4; ++p) {
        const int row = lrow + p * 32;
        const float* ga = aSrc + (size_t)(m0 + row) * 1024 + lcol;
        const float* gb = wSrc + (size_t)(n0 + row) * 1024 + lcol;
        __builtin_amdgcn_global_load_async_to_lds_b128(
            (AS1 v4i*)(v4i*)ga,
            (AS3 v4i*)(v4i*)&AsBuf[row * LSTR + lcol], 0, 0);
        __builtin_amdgcn_global_load_async_to_lds_b128(
            (AS1 v4i*)(v4i*)gb,
            (AS3 v4i*)(v4i*)&BsBuf[row * LSTR + lcol], 0, 0);
    }
#else
    // Fallback: batch ALL global loads first (8 in flight), then LDS stores.
    float4 ra[4], rb[4];
#pragma unroll
    for (int p = 0; p < 4; ++p) {
        const int row = lrow + p * 32;
        ra[p] = *(const float4*)(aSrc + (size_t)(m0 + row) * 1024 + lcol);
        rb[p] = *(const float4*)(wSrc + (size_t)(n0 + row) * 1024 + lcol);
    }
#pragma unroll
    for (int p = 0; p < 4; ++p) {
        const int row = lrow + p * 32;
        *(float4*)&AsBuf[row * LSTR + lcol] = ra[p];
        *(float4*)&BsBuf[row * LSTR + lcol] = rb[p];
    }
#endif
}

__global__ __launch_bounds__(256, 2)
void gate_gemm_f32_wmma(const float* __restrict__ X,    // (B, IN)
                        const float* __restrict__ Hs,   // (B, H)
                        const float* __restrict__ Wx,   // (C, 2H, IN)
                        const float* __restrict__ bx,   // (C, 2H)
                        const float* __restrict__ Wh,   // (C, 2H, H)
                        float* __restrict__ out)        // [tanh block | sigmoid block]
{
    __shared__ float As[2 * TILE_F];
    __shared__ float Bs[2 * TILE_F];

    const int tid    = threadIdx.x;
    const int lane   = tid & 31;
    const int wid    = tid >> 5;       // 0..7
    const int wave_m = wid >> 2;       // 0..1 -> 64 rows each
    const int wave_n = wid & 3;        // 0..3 -> 32 cols each

    const int m0 = blockIdx.x * BM;
    const int n0 = blockIdx.y * BN;

    const int lrow = tid >> 3;         // 0..31
    const int lcol = (tid & 7) * 4;    // 0,4,...,28

    const int lane15 = lane & 15;
    const int khi    = (lane >> 4) * 2;

    v8f acc[4][2];
#pragma unroll
    for (int i = 0; i < 4; ++i)
#pragma unroll
        for (int j = 0; j < 2; ++j) acc[i][j] = (v8f)0.0f;

    // ---- pipeline prologue: stage tile 0 ----
    stage_tiles(0, As, Bs, X, Hs, Wx, Wh, m0, n0, lrow, lcol);
    wait_async_all();
    __syncthreads();

    int buf = 0;
    for (int kt = 0; kt < KDIM; kt += BK) {
        // issue async DMA for the next tile into the other buffer
        if (kt + BK < KDIM)
            stage_tiles(kt + BK, As + (buf ^ 1) * TILE_F, Bs + (buf ^ 1) * TILE_F,
                        X, Hs, Wx, Wh, m0, n0, lrow, lcol);

        const float* As_ = As + buf * TILE_F;
        const float* Bs_ = Bs + buf * TILE_F;

#pragma unroll
        for (int ks = 0; ks < BK / 4; ++ks) {
            v2f aF[4], bF[2];
#pragma unroll
            for (int tm = 0; tm < 4; ++tm) {
                const int r = wave_m * 64 + tm * 16 + lane15;
                aF[tm] = *(const v2f*)&As_[r * LSTR + ks * 4 + khi];
            }
#pragma unroll
            for (int tn = 0; tn < 2; ++tn) {
                const int r = wave_n * 32 + tn * 16 + lane15;
                bF[tn] = *(const v2f*)&Bs_[r * LSTR + ks * 4 + khi];
            }
#pragma unroll
            for (int tm = 0; tm < 4; ++tm)
#pragma unroll
                for (int tn = 0; tn < 2; ++tn)
                    acc[tm][tn] = __builtin_amdgcn_wmma_f32_16x16x4_f32(
                        false, aF[tm], false, bF[tn],
                        (short)0, acc[tm][tn], false, false);
        }

        wait_async_all();     // next tile's DMA done (overlapped with WMMA above)
        __syncthreads();      // all waves done reading buf / writing buf^1
        buf ^= 1;
    }

    // ---- epilogue: bias + activation, non-temporal streaming stores ----
    const int CH = C_DIM * H_DIM;  // 16384
#pragma unroll
    for (int tn = 0; tn < 2; ++tn) {
        const int n = n0 + wave_n * 32 + tn * 16 + lane15;
        const int c = n >> 11;
        const int o = n & 2047;
        const float bias = bx[n];
#pragma unroll
        for (int tm = 0; tm < 4; ++tm) {
            const int mbase = m0 + wave_m * 64 + tm * 16 + ((lane >> 4) << 3);
#pragma unroll
            for (int r = 0; r < 8; ++r) {
                const int m = mbase + r;
                const float v = acc[tm][tn][r] + bias;
                if (o < 1024) {
                    const float res = 1.0f / (1.0f + __expf(-v));   // input_gate
                    __builtin_nontemporal_store(res,
                        &out[(size_t)B_DIM * CH + (size_t)m * CH + (size_t)c * H_DIM + o]);
                } else {
                    const float res = fast_tanh(v);                  // cell_input
                    __builtin_nontemporal_store(res,
                        &out[(size_t)m * CH + (size_t)c * H_DIM + (o - 1024)]);
                }
            }
        }
    }
}

extern "C" void kernel_launch(void* const* d_in, const int* in_sizes, int n_in,
                              void* d_out, int out_size, void* d_ws, size_t ws_size,
                              hipStream_t stream) {
    const float* X  = (const float*)d_in[0];
    const float* Hs = (const float*)d_in[1];
    const float* Wx = (const float*)d_in[2];
    const float* bx = (const float*)d_in[3];
    const float* Wh = (const float*)d_in[4];
    float* out = (float*)d_out;

    dim3 grid(B_DIM / BM, NDIM / BN);   // x = M-tiles fast -> weight tiles L2-resident
    gate_gemm_f32_wmma<<<grid, 256, 0, stream>>>(X, Hs, Wx, bx, Wh, out);
}